// StridedSparseAttention_36206574306108
// MI455X (gfx1250) — compile-verified
//
#include <hip/hip_runtime.h>

// Problem constants (match reference: B=4, S=4096, D=512, STRIDE=8)
#define BATCH 4
#define SEQ   4096
#define DIM   512
#define STRIDE 8
#define NSLOT (SEQ / STRIDE)   // 512 strided key slots per batch
#define QPAD  516              // LDS row pad: bank = (4*m + c) % 64 -> conflict-free col reads
#define QT    32               // queries per workgroup (2 WMMA M-tiles)

typedef float v2f __attribute__((ext_vector_type(2)));
typedef float v4f __attribute__((ext_vector_type(4)));
typedef float v8f __attribute__((ext_vector_type(8)));

// ---------------------------------------------------------------------------
// Kernel 1: compact strided K/V into K-pair-interleaved layouts so each WMMA
// B-fragment (two K values per lane) is a single 8-byte load:
//   kTp[b][k>>1][col][2]  : kTp[(k>>1)*1024 + col*2 + (k&1)] = k[b][8*col][k]
//   vsp[b][s>>1][d][2]    : vsp[(s>>1)*1024 + d*2   + (s&1)] = v[b][8*s][d]
// One block per (batch, slot pair). Source reads are once-only -> NT loads;
// kTp/vsp are re-read by 512 WGs -> regular (L2-resident) stores.
// ---------------------------------------------------------------------------
__global__ __launch_bounds__(256) void stage_kv(
    const float* __restrict__ k, const float* __restrict__ v,
    float* __restrict__ kTp, float* __restrict__ vsp) {
  const int blk = blockIdx.x;           // bb * (NSLOT/2) + sp
  const int bb  = blk >> 8;             // / 256
  const int sp  = blk & 255;            // slot pair
  const int s0  = 2 * sp;
  const size_t srcRow = ((size_t)bb * SEQ + (size_t)s0 * STRIDE) * DIM;
  const float* k0row = k + srcRow;
  const float* k1row = k0row + (size_t)STRIDE * DIM;
  const float* v0row = v + srcRow;
  const float* v1row = v0row + (size_t)STRIDE * DIM;

  v2f* vdst = (v2f*)(vsp + (size_t)bb * NSLOT * DIM) + (size_t)sp * DIM;
  // row sp holds 1024 floats = 512 v2f; element d -> {v[s0][d], v[s1][d]}
  for (int d = threadIdx.x; d < DIM; d += 256) {
    v2f val;
    val.x = __builtin_nontemporal_load(&v0row[d]);
    val.y = __builtin_nontemporal_load(&v1row[d]);
    vdst[d] = val;                       // coalesced b64 stores (RT: reused)
  }

  v2f* kTb = (v2f*)(kTp + (size_t)bb * NSLOT * DIM);
  {
    const int m = threadIdx.x;           // k-pair index 0..255
    v2f p0, p1;
    p0.x = __builtin_nontemporal_load(&k0row[2 * m]);
    p0.y = __builtin_nontemporal_load(&k0row[2 * m + 1]);
    p1.x = __builtin_nontemporal_load(&k1row[2 * m]);
    p1.y = __builtin_nontemporal_load(&k1row[2 * m + 1]);
    kTb[(size_t)m * NSLOT + s0]     = p0;   // one-time scattered 8B writes (4 MB total)
    kTb[(size_t)m * NSLOT + s0 + 1] = p1;
  }
}

// ---------------------------------------------------------------------------
// Kernel 2: mask output [S,S] as float 0/1, float4-vectorized NT stores.
// mask[i][j] = (j % 8 == 0) && (j <= i)
// ---------------------------------------------------------------------------
__global__ __launch_bounds__(256) void write_mask(float* __restrict__ maskOut) {
  const size_t f4 = (size_t)blockIdx.x * 256 + threadIdx.x;  // float4 index
  const int i  = (int)(f4 >> 10);            // / (S/4)
  const int j0 = (int)(f4 & 1023) * 4;       // first column of this float4
  v4f val = {0.f, 0.f, 0.f, 0.f};
  if (((j0 & 7) == 0) && (j0 <= i)) val.x = 1.f;
  __builtin_nontemporal_store(val, &((v4f*)maskOut)[f4]);
}

// ---------------------------------------------------------------------------
// Kernel 3: fused attention. One WG (8 wave32) per (batch, 32-query tile).
// Each B fragment (b64 load) feeds TWO WMMAs (M-tile 0 and 1). All final
// outputs are write-once -> NT stores keep the 8 MB kTp/vsp stream resident
// in the 192 MB L2 against 369 MB of streaming output writes.
// ---------------------------------------------------------------------------
__global__ __launch_bounds__(256) void attn_main(
    const float* __restrict__ q, const float* __restrict__ kTp,
    const float* __restrict__ vsp, float* __restrict__ out,
    float* __restrict__ attn) {
  extern __shared__ __align__(16) float lds[];
  float* qs = lds;                 // [QT][QPAD] query tile
  float* sc = lds + QT * QPAD;     // [QT][QPAD] scores -> weights

  const int blk  = blockIdx.x;     // bb * (SEQ/QT) + qb
  const int bb   = blk >> 7;       // / 128
  const int qb   = blk & 127;
  const int tid  = threadIdx.x;
  const int wave = tid >> 5;
  const int lane = tid & 31;
  const int p    = lane >> 4;      // half-group (K pair select)
  const int hl   = lane & 15;      // M (A frag) / N (B,C frags)

  const float scale = 0.04419417382415922f;  // 1/sqrt(512)

  // ---- stage q tile (QT x 512) into LDS, b128, coalesced, once-read -> NT ----
  {
    const v4f* q4 = (const v4f*)(q + ((size_t)bb * SEQ + (size_t)qb * QT) * DIM);
    for (int it = tid; it < (QT * DIM) / 4; it += 256) {
      const int row = it >> 7;      // / (DIM/4)
      const int c4  = it & 127;
      v4f val = __builtin_nontemporal_load(&q4[row * (DIM / 4) + c4]);
      *(v4f*)(qs + row * QPAD + c4 * 4) = val;
    }
  }
  __syncthreads();

  // ---- GEMM1: scores[QT x 512] = q x kT ----
  const v2f* kTb = (const v2f*)(kTp + (size_t)bb * NSLOT * DIM);  // pair rows of 512 v2f
  for (int ti = 0; ti < 4; ++ti) {
    const int t   = wave + 8 * ti;       // slot tile (0..31)
    const int col = t * 16 + hl;
    v8f c0 = {}, c1 = {};
#pragma unroll 8
    for (int k0 = 0; k0 < DIM; k0 += 4) {
      const int ka = k0 + 2 * p;
      v2f bf = kTb[(size_t)((k0 >> 1) + p) * NSLOT + col];   // single b64 load
      v2f a0, a1;
      a0.x = qs[hl * QPAD + ka];        a0.y = qs[hl * QPAD + ka + 1];
      a1.x = qs[(hl + 16) * QPAD + ka]; a1.y = qs[(hl + 16) * QPAD + ka + 1];
      c0 = __builtin_amdgcn_wmma_f32_16x16x4_f32(false, a0, false, bf,
                                                 (short)0, c0, false, false);
      c1 = __builtin_amdgcn_wmma_f32_16x16x4_f32(false, a1, false, bf,
                                                 (short)0, c1, false, false);
    }
#pragma unroll
    for (int r = 0; r < 8; ++r) {
      sc[(r + 8 * p) * QPAD + t * 16 + hl]      = c0[r] * scale;
      sc[(16 + r + 8 * p) * QPAD + t * 16 + hl] = c1[r] * scale;
    }
  }
  __syncthreads();

  // ---- softmax: wave w handles rows 4w..4w+3; lane owns 16 slots ----
  for (int rr = 0; rr < 4; ++rr) {
    const int m    = wave * 4 + rr;
    const int gq   = qb * QT + m;
    const int smax = gq >> 3;            // last valid slot
    float vals[16];
    float vmax = -3.4e38f;
#pragma unroll
    for (int c0 = 0; c0 < 16; ++c0) {
      const int s = lane + 32 * c0;
      const float x = (s <= smax) ? sc[m * QPAD + s] : -3.4e38f;
      vals[c0] = x;
      vmax = fmaxf(vmax, x);
    }
    for (int off = 16; off > 0; off >>= 1) vmax = fmaxf(vmax, __shfl_xor(vmax, off));
    float sum = 0.f;
#pragma unroll
    for (int c0 = 0; c0 < 16; ++c0) {
      const int s = lane + 32 * c0;
      const float e = (s <= smax) ? __expf(vals[c0] - vmax) : 0.f;
      vals[c0] = e;
      sum += e;
    }
    for (int off = 16; off > 0; off >>= 1) sum += __shfl_xor(sum, off);
    const float inv = 1.f / sum;         // slot 0 always valid -> sum > 0
#pragma unroll
    for (int c0 = 0; c0 < 16; ++c0)
      sc[m * QPAD + lane + 32 * c0] = vals[c0] * inv;
  }
  __syncthreads();

  // ---- write dense attn_weights rows (zeros interleaved), b128 NT stores ----
  {
    v4f* attn4 = (v4f*)(attn + ((size_t)bb * SEQ + (size_t)qb * QT) * SEQ);
    for (int it = tid; it < QT * (SEQ / 4); it += 256) {
      const int row = it >> 10;          // / (SEQ/4)
      const int c4  = it & 1023;
      const int j0  = c4 * 4;
      v4f val = {0.f, 0.f, 0.f, 0.f};
      if ((j0 & 7) == 0) val.x = sc[row * QPAD + (j0 >> 3)];  // masked slots already 0
      __builtin_nontemporal_store(val, &attn4[(size_t)row * (SEQ / 4) + c4]);
    }
  }

  // ---- GEMM2: out[QT x 512] = weights x v_s ----
  const v2f* vsb = (const v2f*)(vsp + (size_t)bb * NSLOT * DIM);  // pair rows of 512 v2f
  float* outb = out + ((size_t)bb * SEQ + (size_t)qb * QT) * DIM;
  for (int ti = 0; ti < 4; ++ti) {
    const int t   = wave + 8 * ti;       // d tile (0..31)
    const int col = t * 16 + hl;
    v8f c0 = {}, c1 = {};
#pragma unroll 8
    for (int k0 = 0; k0 < NSLOT; k0 += 4) {
      const int ka = k0 + 2 * p;
      v2f bf = vsb[(size_t)((k0 >> 1) + p) * DIM + col];     // single b64 load
      v2f a0, a1;
      a0.x = sc[hl * QPAD + ka];        a0.y = sc[hl * QPAD + ka + 1];
      a1.x = sc[(hl + 16) * QPAD + ka]; a1.y = sc[(hl + 16) * QPAD + ka + 1];
      c0 = __builtin_amdgcn_wmma_f32_16x16x4_f32(false, a0, false, bf,
                                                 (short)0, c0, false, false);
      c1 = __builtin_amdgcn_wmma_f32_16x16x4_f32(false, a1, false, bf,
                                                 (short)0, c1, false, false);
    }
#pragma unroll
    for (int r = 0; r < 8; ++r) {
      __builtin_nontemporal_store(c0[r], &outb[(size_t)(r + 8 * p) * DIM + col]);
      __builtin_nontemporal_store(c1[r], &outb[(size_t)(16 + r + 8 * p) * DIM + col]);
    }
  }
}

// ---------------------------------------------------------------------------
extern "C" void kernel_launch(void* const* d_in, const int* in_sizes, int n_in,
                              void* d_out, int out_size, void* d_ws, size_t ws_size,
                              hipStream_t stream) {
  (void)in_sizes; (void)n_in; (void)out_size; (void)ws_size;
  const float* q = (const float*)d_in[0];
  const float* k = (const float*)d_in[1];
  const float* v = (const float*)d_in[2];

  float* out     = (float*)d_out;                                   // [B,S,D]
  float* attn    = out  + (size_t)BATCH * SEQ * DIM;                // [B,S,S]
  float* maskOut = attn + (size_t)BATCH * SEQ * SEQ;                // [S,S]

  float* kTp = (float*)d_ws;                                        // [B][D/2][NSLOT][2]
  float* vsp = kTp + (size_t)BATCH * NSLOT * DIM;                   // [B][NSLOT/2][D][2]

  stage_kv<<<BATCH * (NSLOT / 2), 256, 0, stream>>>(k, v, kTp, vsp);
  write_mask<<<(SEQ * SEQ / 4) / 256, 256, 0, stream>>>(maskOut);

  const size_t ldsBytes = (size_t)2 * QT * QPAD * sizeof(float);    // 132,096 B
  attn_main<<<BATCH * (SEQ / QT), 256, ldsBytes, stream>>>(q, kTp, vsp, out, attn);
}